// TextRelationNTN_61847529063021
// MI455X (gfx1250) — compile-verified
//
#include <hip/hip_runtime.h>

typedef float v2f __attribute__((ext_vector_type(2)));
typedef float v8f __attribute__((ext_vector_type(8)));

constexpr int NB   = 36;    // num features
constexpr int DD   = 256;   // embed size
constexpr int KK   = 8;     // relation slices
constexpr int NPAD = 48;    // 36 padded to 3x16 WMMA tiles
constexpr int XST  = 260;   // Xs row stride (floats): 260 % 64 == 4 -> conflict-free f2 loads
constexpr int AST  = 52;    // As row stride (floats): 2*52 % 64 == 40 -> disjoint half-wave banks

constexpr int SMEM_FLOATS = NPAD * XST   // Xs: X_b zero-padded to 48 rows
                          + DD * AST     // As: A' = W[k] @ X^T, stored [c][n]
                          + NB * NB      // Ts: tanh result for current k, flat n*36+m
                          + NB * NB      // Lg: logits, flat n*36+m
                          + KK * 40      // AddN[k][n] = V1w.x_n + V1b
                          + KK * 40      // AddM[k][m] = (V2w+Wb).x_m + V2b
                          + KK + 4;      // Uw copy + Ub

static __device__ __forceinline__ v8f wmma_f32(v2f a, v2f b, v8f c) {
  // D = A(16x4,f32) * B(4x16,f32) + C(16x16,f32)  -- exact fp32 matrix op on CDNA5
  return __builtin_amdgcn_wmma_f32_16x16x4_f32(false, a, false, b, (short)0, c,
                                               false, false);
}

__global__ __launch_bounds__(256, 1) void ntn_fused(
    const float* __restrict__ texts,  // [B,36,256]
    const float* __restrict__ W,      // [8,256,256]
    const float* __restrict__ Wb,     // [8,256]
    const float* __restrict__ V1w,    // [8,256]
    const float* __restrict__ V1b,    // [8]
    const float* __restrict__ V2w,    // [8,256]
    const float* __restrict__ V2b,    // [8]
    const float* __restrict__ Uw,     // [8]
    const float* __restrict__ Ub,     // [1]
    float* __restrict__ out)          // [B,36,36]
{
  extern __shared__ float smem[];
  float* Xs   = smem;
  float* As   = Xs + NPAD * XST;
  float* Ts   = As + DD * AST;
  float* Lg   = Ts + NB * NB;
  float* AddN = Lg + NB * NB;
  float* AddM = AddN + KK * 40;
  float* Uws  = AddM + KK * 40;
  float* Ubs  = Uws + KK;

  const int tid  = threadIdx.x;
  const int lane = tid & 31;
  const int wave = tid >> 5;
  const int hl   = lane >> 4;   // half-wave select (K/M split in WMMA layouts)
  const int ml   = lane & 15;
  const int b    = blockIdx.x;
  const float* Xg = texts + (size_t)b * NB * DD;

  // ---------------- Phase 0: stage X_b, small rank-1 dots ----------------
  #pragma unroll 4
  for (int n = 0; n < NB; ++n)                      // 256 threads == one row
    Xs[n * XST + tid] = Xg[n * DD + tid];
  for (int i = tid; i < (NPAD - NB) * XST; i += 256)
    Xs[NB * XST + i] = 0.f;                         // zero pad rows 36..47
  if (tid < KK) Uws[tid] = Uw[tid];
  if (tid == 0) Ubs[0] = Ub[0];
  __syncthreads();

  // AddN[k][n] = V1w[k].x_n + V1b[k]; AddM[k][m] = (V2w[k]+Wb[k]).x_m + V2b[k]
  for (int j = tid; j < 2 * KK * NB; j += 256) {
    const int which = j / (KK * NB);
    const int rem   = j - which * (KK * NB);
    const int k = rem / NB, n = rem - k * NB;
    const float* xr = Xs + n * XST;
    if (which == 0) {
      const float* w1 = V1w + k * DD;
      float s = V1b[k];
      for (int d = 0; d < DD; ++d) s += w1[d] * xr[d];
      AddN[k * 40 + n] = s;
    } else {
      const float* w2 = V2w + k * DD;
      const float* wb = Wb + k * DD;
      float s = V2b[k];
      for (int d = 0; d < DD; ++d) s += (w2[d] + wb[d]) * xr[d];
      AddM[k * 40 + n] = s;
    }
  }
  __syncthreads();

  for (int k = 0; k < KK; ++k) {
    // ---- Step 1: A'[c][n] = sum_d W[k,c,d] * X[n,d]  (256x48) ----------
    // wave w owns rows c = 32w..32w+31 (two 16-row tiles), 3 col tiles each
    {
      const float* wk = W + (size_t)k * DD * DD;
      const int c0 = wave * 32;
      v8f a00 = {}, a01 = {}, a02 = {}, a10 = {}, a11 = {}, a12 = {};
      const float* wr0 = wk + (size_t)(c0 + ml) * DD + 2 * hl;       // A rows c0..c0+15
      const float* wr1 = wk + (size_t)(c0 + 16 + ml) * DD + 2 * hl;  // A rows c0+16..+31
      const float* x0  = Xs + (0  + ml) * XST + 2 * hl;              // B = X^T tiles
      const float* x1  = Xs + (16 + ml) * XST + 2 * hl;
      const float* x2  = Xs + (32 + ml) * XST + 2 * hl;
      #pragma unroll 4
      for (int d0 = 0; d0 < DD; d0 += 4) {
        v2f A0 = *(const v2f*)(wr0 + d0);
        v2f A1 = *(const v2f*)(wr1 + d0);
        v2f B0 = *(const v2f*)(x0 + d0);
        v2f B1 = *(const v2f*)(x1 + d0);
        v2f B2 = *(const v2f*)(x2 + d0);
        a00 = wmma_f32(A0, B0, a00);
        a01 = wmma_f32(A0, B1, a01);
        a02 = wmma_f32(A0, B2, a02);
        a10 = wmma_f32(A1, B0, a10);
        a11 = wmma_f32(A1, B1, a11);
        a12 = wmma_f32(A1, B2, a12);
      }
      // C/D layout: vgpr v, lanes 0-15 -> M=v ; lanes 16-31 -> M=8+v
      const int crow0 = c0 + 8 * hl;
      const int crow1 = c0 + 16 + 8 * hl;
      #pragma unroll
      for (int v = 0; v < 8; ++v) {
        As[(crow0 + v) * AST +  0 + ml] = a00[v];
        As[(crow0 + v) * AST + 16 + ml] = a01[v];
        As[(crow0 + v) * AST + 32 + ml] = a02[v];
        As[(crow1 + v) * AST +  0 + ml] = a10[v];
        As[(crow1 + v) * AST + 16 + ml] = a11[v];
        As[(crow1 + v) * AST + 32 + ml] = a12[v];
      }
    }
    __syncthreads();

    // ---- Step 2: F[m][n] = sum_c X[m,c] * A'[c,n]  ( = first[n,m] ) ----
    for (int t = wave; t < 9; t += 8) {             // 3x3 tiles of 16x16
      const int m0 = (t / 3) * 16;
      const int n0 = (t - (t / 3) * 3) * 16;
      v8f acc = {};
      const float* xa = Xs + (m0 + ml) * XST + 2 * hl;  // A = X rows m0..m0+15
      const float* ab = As + n0 + ml;                   // B = A' cols n0..n0+15
      #pragma unroll 4
      for (int cc = 0; cc < DD; cc += 4) {
        v2f A = *(const v2f*)(xa + cc);
        const int cr = cc + 2 * hl;
        v2f Bv;
        Bv.x = ab[(cr + 0) * AST];
        Bv.y = ab[(cr + 1) * AST];
        acc = wmma_f32(A, Bv, acc);
      }
      #pragma unroll
      for (int v = 0; v < 8; ++v) {
        const int m = m0 + v + 8 * hl;
        const int n = n0 + ml;
        if (m < NB && n < NB) {
          // T[k,n,m] = tanh(first[n,m] + Wb.x_m + v1[n] + v2[m])
          Ts[n * NB + m] = tanhf(acc[v] + AddM[k * 40 + m] + AddN[k * 40 + n]);
        }
      }
    }
    __syncthreads();

    // ---- scrambled U-reduction: logits[162k+q] = Ub + dot8(Ts[8q..], Uw)
    for (int q = tid; q < 162; q += 256) {
      float lv = Ubs[0];
      #pragma unroll
      for (int r = 0; r < 8; ++r) lv += Ts[8 * q + r] * Uws[r];
      Lg[162 * k + q] = lv;
    }
    __syncthreads();
  }

  // ---------------- softmax over rows of Lg[36][36] ----------------------
  if (tid < NB) {
    float* row = Lg + tid * NB;
    float mx = row[0];
    for (int m = 1; m < NB; ++m) mx = fmaxf(mx, row[m]);
    float s = 0.f;
    for (int m = 0; m < NB; ++m) { float e = expf(row[m] - mx); row[m] = e; s += e; }
    const float inv = 1.0f / s;
    float* orow = out + ((size_t)b * NB + tid) * NB;
    for (int m = 0; m < NB; ++m) orow[m] = row[m] * inv;
  }
}

extern "C" void kernel_launch(void* const* d_in, const int* in_sizes, int n_in,
                              void* d_out, int out_size, void* d_ws, size_t ws_size,
                              hipStream_t stream) {
  const float* texts = (const float*)d_in[0];
  const float* W     = (const float*)d_in[1];
  const float* Wb    = (const float*)d_in[2];
  const float* V1w   = (const float*)d_in[3];
  const float* V1b   = (const float*)d_in[4];
  const float* V2w   = (const float*)d_in[5];
  const float* V2b   = (const float*)d_in[6];
  const float* Uw    = (const float*)d_in[7];
  const float* Ub    = (const float*)d_in[8];
  float* out = (float*)d_out;

  const int B = in_sizes[0] / (NB * DD);
  const size_t smem = (size_t)SMEM_FLOATS * sizeof(float);
  // allow >64KB dynamic LDS (CDNA5 WGP has 320KB); deterministic, capture-safe
  (void)hipFuncSetAttribute((const void*)ntn_fused,
                            hipFuncAttributeMaxDynamicSharedMemorySize, (int)smem);
  ntn_fused<<<B, 256, smem, stream>>>(texts, W, Wb, V1w, V1b, V2w, V2b, Uw, Ub, out);
}